// CurvatureLoss3D_2757369004564
// MI455X (gfx1250) — compile-verified
//
#include <hip/hip_runtime.h>
#include <hip/hip_bf16.h>
#include <math.h>

// ---------------------------------------------------------------------------
// CurvatureLoss3D on MI455X (gfx1250, wave32).
// - Stencil-as-GEMM: 9 derivative stencils over 3x3x3 neighborhoods as
//   D[16 derivs][16 voxels] += W[16x4] x PHI[4x16] via V_WMMA_F32_16X16X4_F32
//   (full fp32 matrix op, K=27 padded to 28 -> 7 WMMAs per 16-voxel group).
// - HBM -> LDS via CDNA5 async copy (global_load_async_to_lds_b32, ASYNCcnt),
//   no VGPR staging; explicit s_wait_asynccnt before the workgroup barrier.
// ---------------------------------------------------------------------------

typedef __attribute__((ext_vector_type(2))) float v2f;
typedef __attribute__((ext_vector_type(8))) float v8f;

#define VOL      192
#define OUTD     190          // VALID conv output per dim
#define TX       32           // tile outputs along x (contiguous dim)
#define TY       8
#define TZ       4
#define LX       (TX + 2)     // 34
#define LY       (TY + 2)     // 10
#define LZ       (TZ + 2)     // 6
#define LDS_N    (LX * LY * LZ)  // 2040 floats
#define EPSF     1e-8f
#define CURV_T   0.5f

// Stencil weight W[k][n]; k = i0*9 + i1*3 + i2 over the 3x3x3 window,
// n = derivative channel (0..8): gx,gy,gz,hxx,hyy,hzz,hxy,hxz,hyz.
__device__ __forceinline__ float stencil_w(int k, int n) {
    if (k >= 27 || n >= 9) return 0.0f;
    int i0 = k / 9;
    int r  = k - i0 * 9;
    int i1 = r / 3;
    int i2 = r - i1 * 3;
    switch (n) {
        case 0: return (i1 == 1 && i2 == 1) ? (i0 == 0 ? -0.5f : (i0 == 2 ? 0.5f : 0.0f)) : 0.0f;
        case 1: return (i0 == 1 && i2 == 1) ? (i1 == 0 ? -0.5f : (i1 == 2 ? 0.5f : 0.0f)) : 0.0f;
        case 2: return (i0 == 1 && i1 == 1) ? (i2 == 0 ? -0.5f : (i2 == 2 ? 0.5f : 0.0f)) : 0.0f;
        case 3: return (i1 == 1 && i2 == 1) ? (i0 == 1 ? -2.0f : 1.0f) : 0.0f;
        case 4: return (i0 == 1 && i2 == 1) ? (i1 == 1 ? -2.0f : 1.0f) : 0.0f;
        case 5: return (i0 == 1 && i1 == 1) ? (i2 == 1 ? -2.0f : 1.0f) : 0.0f;
        case 6: return (i2 == 1 && i0 != 1 && i1 != 1) ? ((i0 == i1) ? 0.25f : -0.25f) : 0.0f;
        case 7: return (i1 == 1 && i0 != 1 && i2 != 1) ? ((i0 == i2) ? 0.25f : -0.25f) : 0.0f;
        case 8: return (i0 == 1 && i1 != 1 && i2 != 1) ? ((i1 == i2) ? 0.25f : -0.25f) : 0.0f;
    }
    return 0.0f;
}

__device__ __forceinline__ float lds_nb(const float* t, int z, int y, int x, int k) {
    int i0 = k / 9;
    int r  = k - i0 * 9;
    int i1 = r / 3;
    int i2 = r - i1 * 3;
    return t[(z + i0) * (LX * LY) + (y + i1) * LX + (x + i2)];
}

__global__ void __launch_bounds__(256) curv_zero_ws(float* ws) {
    if (threadIdx.x < 2) ws[threadIdx.x] = 0.0f;
}

__global__ void __launch_bounds__(256, 2) curv_main(const float* __restrict__ phi,
                                                    float* __restrict__ ws) {
    __shared__ float tile[LDS_N];

    const int bx    = blockIdx.x;             // 6 tiles of 32 along x
    const int by    = blockIdx.y;             // 24 tiles of 8 along y
    const int bz    = blockIdx.z % 48;        // 48 tiles of 4 along z
    const int batch = blockIdx.z / 48;        // 2 batches

    const int x0 = bx * TX, y0 = by * TY, z0 = bz * TZ;
    const size_t base = (size_t)batch * (size_t)VOL * VOL * VOL;

    // ---- HBM -> LDS via CDNA5 async copy (no VGPR staging, ASYNCcnt) ----
    for (int i = threadIdx.x; i < LDS_N; i += 256) {
        int zz = i / (LX * LY);
        int r  = i - zz * (LX * LY);
        int yy = r / LX;
        int xx = r - yy * LX;
        int gz = min(z0 + zz, VOL - 1);
        int gy = min(y0 + yy, VOL - 1);
        int gx = min(x0 + xx, VOL - 1);
        const float* gptr = phi + base + ((size_t)gz * VOL + gy) * VOL + gx;
        // LDS byte offset = low 32 bits of the generic pointer (LDS aperture).
        unsigned ldsoff = (unsigned)(unsigned long long)(&tile[i]);
        asm volatile("global_load_async_to_lds_b32 %0, %1, off"
                     :: "v"(ldsoff), "v"(gptr)
                     : "memory");
    }
    // Wait for this wave's async global->LDS transfers, then sync the group.
    asm volatile("s_wait_asynccnt 0x0" ::: "memory");
    __syncthreads();

    const int lane = threadIdx.x & 31;
    const int wv   = threadIdx.x >> 5;        // wave id = y row in tile (0..7)
    const int m    = lane & 15;
    const bool hi  = lane >= 16;

    // A-matrix fragments (stencil weights) -- loop invariant, precompute once.
    // f32 16x16x4 A layout: lanes0-15 hold K=kk,kk+1; lanes16-31 hold K=kk+2,kk+3.
    v2f aF[7];
#pragma unroll
    for (int s = 0; s < 7; ++s) {
        int j0 = 4 * s + (hi ? 2 : 0);
        aF[s].x = stencil_w(j0, m);
        aF[s].y = stencil_w(j0 + 1, m);
    }

    float sumP = 0.0f, sumN = 0.0f;

    for (int zr = 0; zr < TZ; ++zr) {
        // ---- 9-channel stencil via matrix pipe: two 16-voxel groups ----
        v8f acc0 = {};
        v8f acc1 = {};
#pragma unroll
        for (int s = 0; s < 7; ++s) {
            int j0 = 4 * s + (hi ? 2 : 0);
            v2f b0, b1;
            b0.x = lds_nb(tile, zr, wv, m, j0);
            b0.y = (j0 + 1 < 27) ? lds_nb(tile, zr, wv, m, j0 + 1) : 0.0f;
            b1.x = lds_nb(tile, zr, wv, 16 + m, j0);
            b1.y = (j0 + 1 < 27) ? lds_nb(tile, zr, wv, 16 + m, j0 + 1) : 0.0f;
            acc0 = __builtin_amdgcn_wmma_f32_16x16x4_f32(false, aF[s], false, b0,
                                                         (short)0, acc0, false, false);
            acc1 = __builtin_amdgcn_wmma_f32_16x16x4_f32(false, aF[s], false, b1,
                                                         (short)0, acc1, false, false);
        }

        // ---- gather: lane l owns voxel x=l with all 9 derivatives ----
        // D layout: VGPR r, lanes0-15 = deriv r, lanes16-31 = deriv 8+r.
        float d[9];
#pragma unroll
        for (int r = 0; r < 8; ++r) {
            float t = __shfl(acc1[r], lane & 15, 32);  // hi lanes fetch from tile1 low lanes
            d[r] = hi ? t : acc0[r];
        }
        {
            float t = __shfl(acc0[0], (lane & 15) | 16, 32);  // deriv8 lives in high lanes
            d[8] = hi ? acc1[0] : t;
        }

        const float gx = d[0], gy = d[1], gzv = d[2];
        const float hxx = d[3], hyy = d[4], hzz = d[5];
        const float hxy = d[6], hxz = d[7], hyz = d[8];

        // ---- zero-crossing min/max over the same 3x3x3 window ----
        float mn = 3.0e38f, mx = -3.0e38f;
#pragma unroll
        for (int i0 = 0; i0 < 3; ++i0)
#pragma unroll
            for (int i1 = 0; i1 < 3; ++i1)
#pragma unroll
                for (int i2 = 0; i2 < 3; ++i2) {
                    float v = tile[(zr + i0) * (LX * LY) + (wv + i1) * LX + (lane + i2)];
                    mn = fminf(mn, v);
                    mx = fmaxf(mx, v);
                }

        // ---- curvature math (fp32 VALU; co-executes with XDL) ----
        float mag2 = gx * gx + gy * gy + gzv * gzv + EPSF;
        float mag  = sqrtf(mag2);
        float mag3 = mag * mag2;
        float cross = gx * gy * hxy + gx * gzv * hxz + gy * gzv * hyz;
        float mean_c = (gx * gx * (hyy + hzz) + gy * gy * (hxx + hzz) +
                        gzv * gzv * (hxx + hyy) - 2.0f * cross) / (mag3 + EPSF);
        float lap  = (hxx + hyy + hzz) / (mag + EPSF);
        float quad = (gx * gx * hxx + gy * gy * hyy + gzv * gzv * hzz + 2.0f * cross) /
                     (mag3 + EPSF);
        float gauss = lap - quad;
        float k1    = mean_c + sqrtf(fabsf(mean_c * mean_c - gauss) + EPSF);
        float ratio = k1 / (CURV_T + EPSF);
        float pen   = fmaxf(ratio * ratio - 1.0f, 0.0f);
        float zc    = (mn * mx < 0.0f) ? 1.0f : 0.0f;

        const int ox = x0 + lane, oy = y0 + wv, oz = z0 + zr;
        if (ox < OUTD && oy < OUTD && oz < OUTD) {
            sumP += zc * pen;
            sumN += zc;
        }
    }

    // ---- wave32 reduction, one atomic pair per wave ----
#pragma unroll
    for (int off = 16; off > 0; off >>= 1) {
        sumP += __shfl_down(sumP, off, 32);
        sumN += __shfl_down(sumN, off, 32);
    }
    if (lane == 0) {
        atomicAdd(&ws[0], sumP);
        atomicAdd(&ws[1], sumN);
    }
}

__global__ void __launch_bounds__(64) curv_finalize(const float* ws, float* out) {
    if (threadIdx.x == 0) out[0] = ws[0] / (ws[1] + EPSF);
}

extern "C" void kernel_launch(void* const* d_in, const int* in_sizes, int n_in,
                              void* d_out, int out_size, void* d_ws, size_t ws_size,
                              hipStream_t stream) {
    (void)in_sizes; (void)n_in; (void)out_size; (void)ws_size;
    const float* phi = (const float*)d_in[0];
    float* out = (float*)d_out;
    float* ws  = (float*)d_ws;

    hipLaunchKernelGGL(curv_zero_ws, dim3(1), dim3(256), 0, stream, ws);
    // 6 x-tiles * 24 y-tiles * (48 z-tiles * 2 batches)
    hipLaunchKernelGGL(curv_main, dim3(6, 24, 96), dim3(256), 0, stream, phi, ws);
    hipLaunchKernelGGL(curv_finalize, dim3(1), dim3(64), 0, stream, ws, out);
}